// MedianFilter_91010357002932
// MI455X (gfx1250) — compile-verified
//
#include <hip/hip_runtime.h>
#include <hip/hip_bf16.h>
#include <stdint.h>

#define IMG_H 512
#define IMG_W 512
#define TS 32
#define PAD 3
#define TW (TS + 2*PAD)     // 38 = tile + halo
#define LSTRIDE 40          // padded LDS row stride (floats)

__device__ __forceinline__ void cswap(float& a, float& b){
  float t = fminf(a, b);
  b = fmaxf(a, b);
  a = t;
}

// Place min of v[0..M-1] in v[0] and max in v[M-1] (values conserved).
// Cost: 3M/2 - 2 CE (even M), 3(M-1)/2 CE (odd M).
template<int M>
__device__ __forceinline__ void minmax_m(float (&v)[26]){
  #pragma unroll
  for (int i = 0; i + 1 < M; i += 2) cswap(v[i], v[i+1]);   // pairwise: lows at even idx
  #pragma unroll
  for (int i = 2; i < M; i += 2) cswap(v[0], v[i]);         // running min -> v[0]
  #pragma unroll
  for (int i = 1; i < M - 1; i += 2) cswap(v[i], v[M-1]);   // running max -> v[M-1]
}

// Forgetful-selection rounds: at step Wd the live set size is m = 52-Wd.
// Drop min (v[0]) and max (v[m-1]); overwrite v[0] with window element Wd.
template<int Wd>
struct Rounds {
  __device__ static __forceinline__ void run(float (&v)[26], const float* p){
    constexpr int m = 52 - Wd;
    minmax_m<m>(v);
    v[0] = p[(Wd/7)*LSTRIDE + (Wd%7)];
    Rounds<Wd+1>::run(v, p);
  }
};
template<>
struct Rounds<49> {
  __device__ static __forceinline__ void run(float (&)[26], const float*){}
};

// Exact median of the 7x7 window rooted at p (LDS), elements p[dy*LSTRIDE+dx].
__device__ __forceinline__ float median49(const float* p){
  float v[26];
  #pragma unroll
  for (int w = 0; w < 26; ++w) v[w] = p[(w/7)*LSTRIDE + (w%7)];
  Rounds<26>::run(v, p);                 // sizes 26..4, all indices compile-time
  cswap(v[0], v[1]); cswap(v[1], v[2]); cswap(v[0], v[1]);   // final 3: middle = median
  return v[1];
}

extern "C" __global__ __launch_bounds__(256)
void median7_wmk(const float* __restrict__ img, float* __restrict__ out)
{
  __shared__ float tile[TW * LSTRIDE];   // 6080 B of the 320 KB WGP LDS

  const int plane = blockIdx.z;          // b*3 + c, 24 planes
  const int c = plane - (plane/3)*3;
  const int gx0 = blockIdx.x * TS;
  const int gy0 = blockIdx.y * TS;
  const float* __restrict__ src = img + (size_t)plane * (IMG_H*IMG_W);
  float* __restrict__ dst = out + (size_t)plane * (IMG_H*IMG_W);

  const float mean = (c==0) ? 0.485f : (c==1) ? 0.456f : 0.406f;
  const float stdv = (c==0) ? 0.229f : (c==1) ? 0.224f : 0.225f;

  const int tid = threadIdx.x;

  // ---- Stage 38x38 reflect-padded tile into LDS via CDNA5 async DMA ----
  // Per-lane global addresses implement jnp.pad(mode='reflect') exactly.
  for (int idx = tid; idx < TW*TW; idx += 256) {
    const int r  = idx / TW;
    const int cc = idx - r*TW;
    int gy = gy0 - PAD + r;
    int gx = gx0 - PAD + cc;
    gy = (gy < 0) ? -gy : ((gy > IMG_H-1) ? (2*IMG_H - 2 - gy) : gy);
    gx = (gx < 0) ? -gx : ((gx > IMG_W-1) ? (2*IMG_W - 2 - gx) : gx);
    const float* gp = src + gy*IMG_W + gx;
    // Low 32 bits of a generic shared pointer == LDS byte offset (aperture in high bits).
    unsigned lds_off = (unsigned)(uintptr_t)&tile[r*LSTRIDE + cc];
    asm volatile("global_load_async_to_lds_b32 %0, %1, off"
                 :: "v"(lds_off), "v"(gp)
                 : "memory");
  }
  asm volatile("s_wait_asynccnt 0" ::: "memory");  // drain this wave's ASYNCcnt
  __syncthreads();                                  // make LDS visible workgroup-wide

  // ---- clip(unnormalize(x), 0, 1) applied once per staged element ----
  for (int idx = tid; idx < TW*TW; idx += 256) {
    const int r  = idx / TW;
    const int cc = idx - r*TW;
    float x = tile[r*LSTRIDE + cc];
    x = fminf(fmaxf(x*stdv + mean, 0.0f), 1.0f);
    tile[r*LSTRIDE + cc] = x;
  }
  __syncthreads();

  // ---- median + renormalize; one wave == one 32-wide row (bank-conflict-free) ----
  const int tx = tid & 31;
  const int ty = tid >> 5;
  const float inv_std = 1.0f / stdv;

  #pragma unroll 1   // keep code size bounded; median49 itself is fully unrolled
  for (int i = 0; i < 4; ++i) {
    const int py = ty*4 + i;
    const float med = median49(&tile[py*LSTRIDE + tx]);
    dst[(size_t)(gy0+py)*IMG_W + (gx0+tx)] = (med - mean) * inv_std;
  }
}

extern "C" __global__ __launch_bounds__(256)
void copy_f4(const float4* __restrict__ in, float4* __restrict__ out, int n)
{
  int i = blockIdx.x * blockDim.x + threadIdx.x;
  if (i < n) out[i] = in[i];
}

extern "C" void kernel_launch(void* const* d_in, const int* in_sizes, int n_in,
                              void* d_out, int out_size, void* d_ws, size_t ws_size,
                              hipStream_t stream) {
  (void)in_sizes; (void)n_in; (void)d_ws; (void)ws_size; (void)out_size;
  const float* img  = (const float*)d_in[0];   // [8,3,512,512] f32
  const float* mask = (const float*)d_in[1];   // [8,1,512,512] f32
  // d_in[2] = kernel_size (always 7; static in the reference)
  float* out = (float*)d_out;

  dim3 grid(IMG_W/TS, IMG_H/TS, 8*3);          // 16 x 16 x 24 tiles
  median7_wmk<<<grid, dim3(256), 0, stream>>>(img, out);

  // mask passthrough -> second output slot
  const int n4 = (8*1*IMG_H*IMG_W) / 4;
  copy_f4<<<(n4 + 255)/256, 256, 0, stream>>>(
      (const float4*)mask, (float4*)(out + (size_t)8*3*IMG_H*IMG_W), n4);
}